// DifferentiableVoxelGrid_88321707474978
// MI455X (gfx1250) — compile-verified
//
#include <hip/hip_runtime.h>
#include <hip/hip_bf16.h>

typedef __attribute__((ext_vector_type(2))) float v2f;
typedef __attribute__((ext_vector_type(8))) float v8f;

#define IMG   512
#define HW    (IMG * IMG)        // 262144
#define NVOX  (128 * 128 * 128)  // 2097152 == 2^21

// ---------------------------------------------------------------------------
// Pass 0: reset the per-pixel min-key z-buffer (lives in d_ws, L2-resident 2MB)
// ---------------------------------------------------------------------------
__global__ __launch_bounds__(256) void vx_init(ulonglong2* __restrict__ mk2) {
  int p = blockIdx.x * 256 + threadIdx.x;
  mk2[p] = make_ulonglong2(~0ULL, ~0ULL);
}

// ---------------------------------------------------------------------------
// Pass 1: project voxels (4 per thread via b128 occupancy load), cull,
// atomic-min 64-bit key into z-buffer.
// key = (bitcast(w) << 21) | voxel_index   (N == 2^21, so index fits exactly)
// Invalid voxels are skipped: their reference keys carry inf depth bits and
// can never win, so the output is identical.
// ---------------------------------------------------------------------------
__global__ __launch_bounds__(256) void vx_project(const float* __restrict__ occ_logits,
                                                  const float* __restrict__ view,
                                                  const float* __restrict__ proj,
                                                  unsigned long long* __restrict__ mk) {
  __shared__ float P[16];  // P = proj @ view, computed once per block
  int tid = threadIdx.x;
  if (tid < 16) {
    int r = tid >> 2, c = tid & 3;
    float acc = proj[r * 4 + 0] * view[0 + c];
    acc += proj[r * 4 + 1] * view[4 + c];
    acc += proj[r * 4 + 2] * view[8 + c];
    acc += proj[r * 4 + 3] * view[12 + c];
    P[tid] = acc;
  }
  __syncthreads();

  int t = blockIdx.x * 256 + tid;
  int base = t * 4;  // 4 consecutive voxels per thread (same i, j; k = k0..k0+3)
  float4 xo4 = ((const float4*)occ_logits)[t];  // global_load_b128
  float xs[4] = {xo4.x, xo4.y, xo4.z, xo4.w};

  float Pr[16];
#pragma unroll
  for (int e = 0; e < 16; ++e) Pr[e] = P[e];

  int i = base >> 14;         // X index (Y*Z = 2^14)
  int j = (base >> 7) & 127;  // Y index
  int k0 = base & 127;        // Z index of first voxel (multiple of 4, +3 <= 127)
  float wx = (((float)i + 0.5f) * (1.0f / 128.0f) * 2.0f - 1.0f) * 2.0f;
  float wy = (((float)j + 0.5f) * (1.0f / 128.0f) * 2.0f - 1.0f) * 2.0f;

  // First two summation terms are shared by all 4 voxels of this thread;
  // per-voxel completion below preserves the reference's rounding order.
  float tx = Pr[0]  * wx + Pr[1]  * wy;
  float ty = Pr[4]  * wx + Pr[5]  * wy;
  float tz = Pr[8]  * wx + Pr[9]  * wy;
  float tw = Pr[12] * wx + Pr[13] * wy;

#pragma unroll
  for (int q = 0; q < 4; ++q) {
    float occ = 1.0f / (1.0f + expf(-xs[q]));
    if (!(occ > 0.01f)) continue;  // OCC_THRESHOLD

    int n = base + q;
    float wz = (((float)(k0 + q) + 0.5f) * (1.0f / 128.0f) * 2.0f - 1.0f) * 2.0f;

    float cx = (tx + Pr[2]  * wz) + Pr[3];
    float cy = (ty + Pr[6]  * wz) + Pr[7];
    float cz = (tz + Pr[10] * wz) + Pr[11];
    float cw = (tw + Pr[14] * wz) + Pr[15];
    if (!(cw > 1e-6f)) continue;

    float nx = cx / cw, ny = cy / cw, nz = cz / cw;
    float px = (nx + 1.0f) * 0.5f * (float)IMG;
    float py = (1.0f - ny) * 0.5f * (float)IMG;
    if (!(px >= 0.0f && px < (float)IMG && py >= 0.0f && py < (float)IMG &&
          nz > -1.0f && nz < 1.0f))
      continue;

    int pxi = min(max((int)px, 0), IMG - 1);
    int pyi = min(max((int)py, 0), IMG - 1);
    int pix = pyi * IMG + pxi;

    unsigned long long key =
        ((unsigned long long)__float_as_uint(cw) << 21) | (unsigned long long)(unsigned)n;
    atomicMin(&mk[pix], key);  // -> global_atomic_min_u64
  }
}

// ---------------------------------------------------------------------------
// Pass 2: pixel-centric shade. Decode winner voxel from key, gather its 8
// material logits with two b128 loads, softmax, then batch the
// (probs x material_colors) einsum through V_WMMA_F32_16X16X4_F32 (two
// 16-pixel groups per wave, K=8 as two chained K=4 WMMAs). Composite with
// sky and write (1,4,H,W).
// ---------------------------------------------------------------------------
__global__ __launch_bounds__(256) void vx_shade(const float* __restrict__ occ_logits,
                                                const float* __restrict__ mat_logits,
                                                const float* __restrict__ mat_colors,
                                                const unsigned long long* __restrict__ mk,
                                                float* __restrict__ out) {
  __shared__ float probsS[256 * 9];  // stride 9 to spread LDS banks
  __shared__ float colS[256 * 3];

  int tid = threadIdx.x;
  int lane = tid & 31;
  int waveBase = tid & ~31;
  int p = blockIdx.x * 256 + tid;

  unsigned long long key = mk[p];
  bool hit = (key != ~0ULL);

  float occ = 0.0f;
  float pr[8];
#pragma unroll
  for (int m = 0; m < 8; ++m) pr[m] = 0.0f;

  if (hit) {
    unsigned vox = (unsigned)(key & 0x1FFFFFULL);
    float xo = occ_logits[vox];
    occ = 1.0f / (1.0f + expf(-xo));
    // 32B-aligned contiguous row -> two global_load_b128
    const float4* ml4 = (const float4*)(mat_logits + (size_t)vox * 8);
    float4 mla = ml4[0];
    float4 mlb = ml4[1];
    float ml[8] = {mla.x, mla.y, mla.z, mla.w, mlb.x, mlb.y, mlb.z, mlb.w};
    float mod[8], mx = -INFINITY;
#pragma unroll
    for (int m = 0; m < 8; ++m) {
      mod[m] = ml[m] * occ;
      mx = fmaxf(mx, mod[m]);
    }
    float s = 0.0f;
#pragma unroll
    for (int m = 0; m < 8; ++m) {
      pr[m] = expf(mod[m] - mx);
      s += pr[m];
    }
#pragma unroll
    for (int m = 0; m < 8; ++m) pr[m] = pr[m] / s;
  }

#pragma unroll
  for (int m = 0; m < 8; ++m) probsS[tid * 9 + m] = pr[m];
  __syncthreads();

  // B fragments: B[K][N] = material_colors[kb+K][N] (N<3, zero-padded to 16).
  // f32 4x16 B layout: lane n holds column n; lanes 0-15 K=0/1, lanes 16-31 K=2/3.
  int ncol = lane & 15;
  int koff = (lane & 16) ? 2 : 0;
  v2f b0, b1;
  b0.x = (ncol < 3) ? mat_colors[(0 + koff) * 3 + ncol] : 0.0f;
  b0.y = (ncol < 3) ? mat_colors[(1 + koff) * 3 + ncol] : 0.0f;
  b1.x = (ncol < 3) ? mat_colors[(4 + koff) * 3 + ncol] : 0.0f;
  b1.y = (ncol < 3) ? mat_colors[(5 + koff) * 3 + ncol] : 0.0f;

#pragma unroll
  for (int g = 0; g < 2; ++g) {
    // A fragments: f32 16x4 layout: lanes 0-15 (M=lane) K=0/1, lanes 16-31 K=2/3.
    int row = waveBase + g * 16 + ncol;
    v2f a0, a1;
    a0.x = probsS[row * 9 + koff + 0];
    a0.y = probsS[row * 9 + koff + 1];
    a1.x = probsS[row * 9 + 4 + koff + 0];
    a1.y = probsS[row * 9 + 4 + koff + 1];

    v8f c = {0.f, 0.f, 0.f, 0.f, 0.f, 0.f, 0.f, 0.f};
    c = __builtin_amdgcn_wmma_f32_16x16x4_f32(false, a0, false, b0, (short)0, c, false, false);
    c = __builtin_amdgcn_wmma_f32_16x16x4_f32(false, a1, false, b1, (short)0, c, false, false);

    // D layout: VGPR r -> (M=r, N=lane) for lanes 0-15; (M=r+8, N=lane-16) for 16-31.
    if (ncol < 3) {
      int rbase = waveBase + g * 16 + ((lane & 16) ? 8 : 0);
#pragma unroll
      for (int r = 0; r < 8; ++r) colS[(rbase + r) * 3 + ncol] = c[r];
    }
  }
  __syncthreads();

  float c0 = colS[tid * 3 + 0];
  float c1 = colS[tid * 3 + 1];
  float c2 = colS[tid * 3 + 2];

  float rr, gg, bb, aa;
  if (hit) {
    float s1 = 1.0f - occ;
    rr = occ * c0 + s1 * 0.5f;
    gg = occ * c1 + s1 * 0.7f;
    bb = occ * c2 + s1 * 0.9f;
    aa = occ;
  } else {
    rr = 0.5f; gg = 0.7f; bb = 0.9f; aa = 0.0f;
  }
  out[0 * HW + p] = rr;
  out[1 * HW + p] = gg;
  out[2 * HW + p] = bb;
  out[3 * HW + p] = aa;
}

extern "C" void kernel_launch(void* const* d_in, const int* in_sizes, int n_in,
                              void* d_out, int out_size, void* d_ws, size_t ws_size,
                              hipStream_t stream) {
  const float* occ  = (const float*)d_in[0];  // occupancy_logits (128^3)
  const float* mat  = (const float*)d_in[1];  // material_logits  (128^3 x 8)
  const float* view = (const float*)d_in[2];  // camera_view (4x4)
  const float* proj = (const float*)d_in[3];  // camera_proj (4x4)
  const float* mcol = (const float*)d_in[4];  // material_colors (8x3)
  (void)in_sizes; (void)n_in; (void)out_size; (void)ws_size;

  unsigned long long* mk = (unsigned long long*)d_ws;  // 2 MB z-key buffer
  float* out = (float*)d_out;                          // (1,4,512,512)

  vx_init<<<HW / 512, 256, 0, stream>>>((ulonglong2*)mk);
  vx_project<<<NVOX / 1024, 256, 0, stream>>>(occ, view, proj, mk);
  vx_shade<<<HW / 256, 256, 0, stream>>>(occ, mat, mcol, mk, out);
}